// Pointer_26113401160110
// MI455X (gfx1250) — compile-verified
//
#include <hip/hip_runtime.h>
#include <math.h>

// ---------------------------------------------------------------------------
// Additive attention:  logits[b,q,k] = sum_d tanh(key[b,k,d]+query[b,q,d])*v_w[d] + v_b
// key = enc @ W1 + b1   (WMMA f32 16x16x4, exact fp32)
// query = in @ W2 + b2  (WMMA f32 16x16x4, exact fp32)
// Scores: LDS-tiled tanh-dot with async double-buffered key tiles
//         (global_load_async_to_lds_b128 + s_wait_asynccnt), fused argmax.
// ---------------------------------------------------------------------------

#define BB 4
#define SS 512
#define DD 128
#define MTOT (BB * SS)          // 2048 rows per projection

typedef __attribute__((ext_vector_type(2)))  float v2f;
typedef __attribute__((ext_vector_type(8)))  float v8f;

#if __has_builtin(__builtin_amdgcn_tanhf)
  #define TANHF(x) __builtin_amdgcn_tanhf(x)
#elif __has_builtin(__builtin_amdgcn_tanh_f32)
  #define TANHF(x) __builtin_amdgcn_tanh_f32(x)
#else
  #define TANHF(x) tanhf(x)
#endif

// ---------------- Projection GEMM: out[m,n] = sum_k X[m,k]*W[k,n] + bias[n] ----
// One wave (32 lanes) computes one 16x16 output tile. K = 128.
__global__ __launch_bounds__(32) void proj_kernel(const float* __restrict__ X,
                                                  const float* __restrict__ W,
                                                  const float* __restrict__ bias,
                                                  float* __restrict__ out) {
  const int tm   = blockIdx.x;          // 0..127 (M tiles of 16)
  const int tn   = blockIdx.y;          // 0..7   (N tiles of 16)
  const int lane = threadIdx.x;         // wave32
  const int r    = lane & 15;
  const int half = lane >> 4;

  // A 16x4: lane r = row M, K = 2*half + vgpr  -> two consecutive K per lane.
  // B 4x16: lane r = col N, K = 2*half + vgpr.
  const float* arow = X + (tm * 16 + r) * DD + half * 2;
  const float* bcol = W + (half * 2) * DD + tn * 16 + r;
  v8f acc = {};
#pragma unroll 8
  for (int k0 = 0; k0 < DD; k0 += 4) {
    v2f a; a[0] = arow[k0];           a[1] = arow[k0 + 1];
    v2f b; b[0] = bcol[k0 * DD];      b[1] = bcol[k0 * DD + DD];
    acc = __builtin_amdgcn_wmma_f32_16x16x4_f32(false, a, false, b,
                                                (short)0, acc, false, false);
  }

  // D layout: vgpr i, lane -> row = tm*16 + i + 8*half, col = tn*16 + r
  const float bv = bias[tn * 16 + r];
  float* orow = out + (tm * 16 + 8 * half) * DD + tn * 16 + r;
#pragma unroll
  for (int i = 0; i < 8; ++i) orow[i * DD] = acc[i] + bv;
}

// ---------------- Attention score + mask + fused argmax ----------------------
// Block = 256 threads; owns (b, 16 queries), streams all 512 keys in 16-row
// tiles via async DMA into double-buffered LDS.
#define QT 16
#define KT 16
#define NKT (SS / KT)   // 32 key tiles
#define LSTR 132        // 128 + 4 pad: kills 16-way LDS bank conflicts (132 % 64 == 4)

// Issue this thread's share (2 x b128) of one 16x128 key tile as async
// global->LDS DMA.  Per wave: 2 instructions x 32 lanes x 16B = 1KB.
__device__ __forceinline__ void async_tile_issue(float* dst_lds, const float* src_glb,
                                                 int t) {
#pragma unroll
  for (int i = t; i < (KT * DD) / 4; i += 256) {   // 512 chunks of 16B
    const int row = i >> 5;                         // 0..15
    const int c4  = (i & 31) << 2;                  // col (floats), 16B aligned
    // Low 32 bits of a generic LDS pointer == LDS byte offset (ISA 10.2).
    unsigned int lds = (unsigned int)(size_t)(const void*)(dst_lds + row * LSTR + c4);
    const float* g = src_glb + row * DD + c4;
    asm volatile("global_load_async_to_lds_b128 %0, %1, off"
                 :: "v"(lds), "v"(g) : "memory");
  }
}

__global__ __launch_bounds__(256) void attn_kernel(const float* __restrict__ key,
                                                   const float* __restrict__ query,
                                                   const int*   __restrict__ mask,
                                                   const float* __restrict__ vw,
                                                   const float* __restrict__ vb,
                                                   float* __restrict__ logits,
                                                   float* __restrict__ preds) {
  __shared__ __align__(16) float qs[QT * LSTR];
  __shared__ __align__(16) float ks[2][KT * LSTR];   // double buffer
  __shared__ __align__(16) float vws[DD];
  __shared__ float redv[QT][KT];
  __shared__ int   redi[QT][KT];

  const int t  = threadIdx.x;
  const int b  = blockIdx.x >> 5;          // / 32
  const int q0 = (blockIdx.x & 31) << 4;   // * 16

  // Stage 16 query rows + v_w into LDS (coalesced, one-time).
  const float* qbase = query + (b * SS + q0) * DD;
  for (int i = t; i < QT * DD; i += 256) {
    qs[(i >> 7) * LSTR + (i & 127)] = qbase[i];
  }
  if (t < DD) vws[t] = vw[t];
  const float vbv = vb[0];

  const float* kbatch = key + b * SS * DD;

  // Prologue: start DMA of key tile 0 into buffer 0.
  async_tile_issue(ks[0], kbatch, t);

  const int ql = t >> 4;    // 0..15 query within tile
  const int kl = t & 15;    // 0..15 key within tile
  float best  = -INFINITY;
  int   besti = 0x7fffffff;

  for (int kt = 0; kt < NKT; ++kt) {
    __syncthreads();  // prior tile's consumers done -> safe to overwrite other buf
    if (kt + 1 < NKT) {
      async_tile_issue(ks[(kt + 1) & 1], kbatch + (kt + 1) * KT * DD, t);
      // Async loads complete in order: <=2 outstanding means this wave's
      // 2 transfers for tile kt have landed; tile kt+1's remain in flight.
      asm volatile("s_wait_asynccnt 0x2" ::: "memory");
    } else {
      asm volatile("s_wait_asynccnt 0x0" ::: "memory");
    }
    __syncthreads();  // every wave's share of tile kt is resident

    const int kg = kt * KT + kl;
    const float* qr = &qs[ql * LSTR];
    const float* kr = &ks[kt & 1][kl * LSTR];
    float sum = 0.f;
#pragma unroll
    for (int d = 0; d < DD; d += 4) {
      float4 qv = *(const float4*)(qr + d);
      float4 kv = *(const float4*)(kr + d);
      float4 vv = *(const float4*)(vws + d);
      sum += TANHF(qv.x + kv.x) * vv.x;
      sum += TANHF(qv.y + kv.y) * vv.y;
      sum += TANHF(qv.z + kv.z) * vv.z;
      sum += TANHF(qv.w + kv.w) * vv.w;
    }
    const float ui = sum + vbv;
    const float lg = (mask[b * SS + kg] != 0) ? ui : -INFINITY;
    logits[(b * SS + q0 + ql) * SS + kg] = lg;
    if (lg > best || (lg == best && kg < besti)) { best = lg; besti = kg; }
  }

  // Cross-thread argmax per query (16 threads share a q), first-index wins.
  redv[ql][kl] = best;
  redi[ql][kl] = besti;
  __syncthreads();
  if (kl == 0) {
    float bvv = redv[ql][0];
    int   bii = redi[ql][0];
#pragma unroll
    for (int j = 1; j < KT; ++j) {
      float v = redv[ql][j]; int i2 = redi[ql][j];
      if (v > bvv || (v == bvv && i2 < bii)) { bvv = v; bii = i2; }
    }
    preds[b * SS + q0 + ql] = (float)bii;
  }
}

// ---------------------------------------------------------------------------
extern "C" void kernel_launch(void* const* d_in, const int* in_sizes, int n_in,
                              void* d_out, int out_size, void* d_ws, size_t ws_size,
                              hipStream_t stream) {
  const float* inputs = (const float*)d_in[0];  // [B,S,D]  -> query via W2
  const float* enc    = (const float*)d_in[1];  // [B,S,D]  -> key via W1
  const int*   mask   = (const int*)  d_in[2];  // [B,S]
  const float* W1     = (const float*)d_in[3];
  const float* b1     = (const float*)d_in[4];
  const float* W2     = (const float*)d_in[5];
  const float* b2     = (const float*)d_in[6];
  const float* v_w    = (const float*)d_in[7];
  const float* v_b    = (const float*)d_in[8];

  float* out       = (float*)d_out;
  float* logits    = out;                              // B*S*S floats
  float* preds     = out + (size_t)BB * SS * SS;       // B*S floats
  float* key_out   = (float*)d_ws;                     // MTOT*DD floats
  float* query_out = key_out + (size_t)MTOT * DD;      // MTOT*DD floats

  dim3 gproj(MTOT / 16, DD / 16);                      // (128, 8)
  proj_kernel<<<gproj, 32, 0, stream>>>(enc,    W1, b1, key_out);
  proj_kernel<<<gproj, 32, 0, stream>>>(inputs, W2, b2, query_out);

  attn_kernel<<<(BB * SS) / QT, 256, 0, stream>>>(key_out, query_out, mask,
                                                  v_w, v_b, logits, preds);
}